// GCN_5385888989316
// MI455X (gfx1250) — compile-verified
//
#include <hip/hip_runtime.h>
#include <hip/hip_bf16.h>

// ---------------------------------------------------------------------------
// 2-layer GCN (eval):  out = Agg( relu(BN( Agg(x@W1) + b1 )) @ W2 ) + b2
//   Agg = D^-1/2 (A+I) D^-1/2, messages src->dst.
// GEMMs: CDNA5 V_WMMA_F32_16X16X4_F32, weights staged in LDS (W1 via TDM
// tensor_load_to_lds). BN+ReLU fused into GEMM2 A-loads; self-loop terms
// fused into GEMM epilogues. Edge scatter: hardware global_atomic_add_f32.
// ---------------------------------------------------------------------------

typedef __attribute__((ext_vector_type(2))) float v2f;
typedef __attribute__((ext_vector_type(8))) float v8f;
typedef __attribute__((ext_vector_type(4))) unsigned int v4u;
typedef __attribute__((ext_vector_type(8))) int v8i_;
typedef __attribute__((ext_vector_type(4))) int v4i_;

#define BN_EPS 1e-5f

// ---------------- degree / norm ----------------
__global__ void k_init_deg(float* __restrict__ deg, int N) {
    int i = blockIdx.x * blockDim.x + threadIdx.x;
    if (i < N) deg[i] = 1.0f;                       // self loop
}

__global__ void k_count_deg(const int* __restrict__ dst, float* __restrict__ deg, int E) {
    int e = blockIdx.x * blockDim.x + threadIdx.x;
    if (e < E) unsafeAtomicAdd(&deg[dst[e]], 1.0f);
}

__global__ void k_rsqrt_inplace(float* __restrict__ deg, int N) {
    int i = blockIdx.x * blockDim.x + threadIdx.x;
    if (i < N) deg[i] = rsqrtf(deg[i]);             // deg >= 1 always
}

// ---------------- BN folding: scale[c], shift[c] ----------------
__global__ void k_prep_bn(const float* __restrict__ b1, const float* __restrict__ gamma,
                          const float* __restrict__ beta, const float* __restrict__ mean,
                          const float* __restrict__ var,
                          float* __restrict__ scale, float* __restrict__ shift) {
    int c = threadIdx.x;
    if (c < 128) {
        float s = gamma[c] * rsqrtf(var[c] + BN_EPS);
        scale[c] = s;
        shift[c] = beta[c] + (b1[c] - mean[c]) * s;
    }
}

// ---------------- GEMM1: G1 = X @ W1 ; A1 = dinv^2 * G1 (self-loop seed) ---
// Block = 256 threads (8 waves); W1 (128x128 f32 = 64KB) staged in LDS via
// the Tensor Data Mover. Each wave computes a 16-row x 128-col stripe.
__global__ __launch_bounds__(256) void k_gemm1(const float* __restrict__ X,
                                               const float* __restrict__ W,
                                               const float* __restrict__ dinv,
                                               float* __restrict__ G,
                                               float* __restrict__ A1, int N) {
    __shared__ float sW[128 * 128];                 // 64 KB (LDS is 320KB/WGP)

    // --- TDM: DMA W1 (contiguous 128x128 f32) into LDS; one wave issues ---
    if (threadIdx.x < 32) {
        unsigned long long ga = (unsigned long long)(const void*)W;
        v4u g0;
        g0.x = 1u;                                          // count=1 (valid)
        g0.y = (unsigned)(unsigned long long)(const void*)sW; // lds byte offset
        g0.z = (unsigned)(ga & 0xFFFFFFFFull);              // global_addr[31:0]
        g0.w = (unsigned)((ga >> 32) & 0x1FFFFFFull) | (2u << 30); // [56:32] | type=2
        v8i_ g1d;
        g1d[0] = (int)(2u << 16);       // wg_mask=0, data_size=2 (4B)
        g1d[1] = (int)(128u << 16);     // tensor_dim0 = 128 (low 16 in [31:16])
        g1d[2] = (int)(128u << 16);     // tensor_dim0 hi=0 ; tensor_dim1 = 128
        g1d[3] = (int)(128u << 16);     // tensor_dim1 hi=0 ; tile_dim0 = 128
        g1d[4] = (int)(128u);           // tile_dim1 = 128 ; tile_dim2 = 0
        g1d[5] = (int)(128u);           // tensor_dim0_stride = 128 (low 32)
        g1d[6] = 0;                     // stride hi / dim1_stride lo
        g1d[7] = 0;
        v4i_ z4 = {};
        v8i_ z8 = {};
        __builtin_amdgcn_tensor_load_to_lds(g0, g1d, z4, z4, z8, 0);
        __builtin_amdgcn_s_wait_tensorcnt(0);
    }
    __syncthreads();

    const int lane = threadIdx.x & 31;
    const int wave = threadIdx.x >> 5;
    const int m0 = (blockIdx.x * 8 + wave) * 16;
    if (m0 < N) {
        const int col = lane & 15;
        const int kof = (lane >> 4) * 2;
        int mrow = m0 + col;
        if (mrow >= N) mrow = N - 1;                // clamp (loads only)

        v8f acc[8];
        #pragma unroll
        for (int j = 0; j < 8; ++j) { v8f z = {}; acc[j] = z; }

        for (int k0 = 0; k0 < 128; k0 += 4) {
            const v2f a = *(const v2f*)(X + (size_t)mrow * 128 + k0 + kof);
            #pragma unroll
            for (int j = 0; j < 8; ++j) {
                const float* bp = sW + (k0 + kof) * 128 + j * 16 + col;
                v2f b; b.x = bp[0]; b.y = bp[128];
                acc[j] = __builtin_amdgcn_wmma_f32_16x16x4_f32(
                    false, a, false, b, (short)0, acc[j], false, false);
            }
        }

        const int srow = (lane >> 4) * 8;
        #pragma unroll
        for (int r = 0; r < 8; ++r) {
            int m = m0 + srow + r;
            if (m < N) {
                float di = dinv[m];
                float di2 = di * di;
                #pragma unroll
                for (int j = 0; j < 8; ++j) {
                    float v = acc[j][r];
                    size_t o = (size_t)m * 128 + j * 16 + col;
                    G[o] = v;
                    A1[o] = di2 * v;                // seed = self-loop term
                }
            }
        }
    }
}

// ---------------- GEMM2: H = relu(A1*scale+shift) on the fly;
//                  G2 = H @ W2 ; out = b2 + dinv^2 * G2 -----------------
__global__ __launch_bounds__(256) void k_gemm2(const float* __restrict__ A1raw,
                                               const float* __restrict__ W2,
                                               const float* __restrict__ dinv,
                                               const float* __restrict__ scale,
                                               const float* __restrict__ shift,
                                               const float* __restrict__ b2,
                                               float* __restrict__ G2,
                                               float* __restrict__ out, int N) {
    __shared__ float sW[128 * 48];                  // W2 zero-padded 40->48
    __shared__ float sScale[128];
    __shared__ float sShift[128];
    __shared__ float sB2[48];

    for (int i = threadIdx.x; i < 128 * 48; i += 256) {
        int k = i / 48, n = i - k * 48;
        sW[i] = (n < 40) ? W2[k * 40 + n] : 0.0f;
    }
    if (threadIdx.x < 128) {
        sScale[threadIdx.x] = scale[threadIdx.x];
        sShift[threadIdx.x] = shift[threadIdx.x];
    }
    if (threadIdx.x < 48) sB2[threadIdx.x] = (threadIdx.x < 40) ? b2[threadIdx.x] : 0.0f;
    __syncthreads();

    const int lane = threadIdx.x & 31;
    const int wave = threadIdx.x >> 5;
    const int m0 = (blockIdx.x * 8 + wave) * 16;
    if (m0 < N) {
        const int col = lane & 15;
        const int kof = (lane >> 4) * 2;
        int mrow = m0 + col;
        if (mrow >= N) mrow = N - 1;

        v8f acc[3];
        #pragma unroll
        for (int j = 0; j < 3; ++j) { v8f z = {}; acc[j] = z; }

        for (int k0 = 0; k0 < 128; k0 += 4) {
            const int c0 = k0 + kof;
            v2f a = *(const v2f*)(A1raw + (size_t)mrow * 128 + c0);
            float ax = a.x * sScale[c0] + sShift[c0];       // fused bias+BN
            float ay = a.y * sScale[c0 + 1] + sShift[c0 + 1];
            a.x = ax > 0.0f ? ax : 0.0f;                    // fused ReLU
            a.y = ay > 0.0f ? ay : 0.0f;
            #pragma unroll
            for (int j = 0; j < 3; ++j) {
                const float* bp = sW + c0 * 48 + j * 16 + col;
                v2f b; b.x = bp[0]; b.y = bp[48];
                acc[j] = __builtin_amdgcn_wmma_f32_16x16x4_f32(
                    false, a, false, b, (short)0, acc[j], false, false);
            }
        }

        const int srow = (lane >> 4) * 8;
        #pragma unroll
        for (int r = 0; r < 8; ++r) {
            int m = m0 + srow + r;
            if (m < N) {
                float di = dinv[m];
                float di2 = di * di;
                #pragma unroll
                for (int j = 0; j < 3; ++j) {
                    int c = j * 16 + col;
                    if (c < 40) {
                        float v = acc[j][r];
                        size_t o = (size_t)m * 40 + c;
                        G2[o] = v;
                        out[o] = sB2[c] + di2 * v;  // seed = b2 + self loop
                    }
                }
            }
        }
    }
}

// ---------------- edge aggregation, layer 1 (128 cols, wave per edge) ------
__global__ __launch_bounds__(256) void k_edge1(const int* __restrict__ src,
                                               const int* __restrict__ dst,
                                               const float* __restrict__ dinv,
                                               const float* __restrict__ G1,
                                               float* __restrict__ A1, int E) {
    int e = (int)((blockIdx.x * blockDim.x + threadIdx.x) >> 5);
    int lane = threadIdx.x & 31;
    if (e >= E) return;
    int s = src[e], d = dst[e];
    float w = dinv[s] * dinv[d];
    const float4 v = *(const float4*)(G1 + (size_t)s * 128 + lane * 4);
    float* o = A1 + (size_t)d * 128 + lane * 4;
    unsafeAtomicAdd(o + 0, w * v.x);
    unsafeAtomicAdd(o + 1, w * v.y);
    unsafeAtomicAdd(o + 2, w * v.z);
    unsafeAtomicAdd(o + 3, w * v.w);
}

// ---------------- edge aggregation, layer 2 (40 cols, wave per edge) -------
__global__ __launch_bounds__(256) void k_edge2(const int* __restrict__ src,
                                               const int* __restrict__ dst,
                                               const float* __restrict__ dinv,
                                               const float* __restrict__ G2,
                                               float* __restrict__ out, int E) {
    int e = (int)((blockIdx.x * blockDim.x + threadIdx.x) >> 5);
    int lane = threadIdx.x & 31;
    if (e >= E) return;
    int s = src[e], d = dst[e];
    float w = dinv[s] * dinv[d];
    const float* g = G2 + (size_t)s * 40;
    float* o = out + (size_t)d * 40;
    unsafeAtomicAdd(o + lane, w * g[lane]);                          // cols 0..31
    if (lane < 8) unsafeAtomicAdd(o + 32 + lane, w * g[32 + lane]);  // cols 32..39
}

// ---------------------------------------------------------------------------
extern "C" void kernel_launch(void* const* d_in, const int* in_sizes, int n_in,
                              void* d_out, int out_size, void* d_ws, size_t ws_size,
                              hipStream_t stream) {
    const float* x     = (const float*)d_in[0];
    const int*   eidx  = (const int*)  d_in[1];
    const float* W1    = (const float*)d_in[2];
    const float* b1    = (const float*)d_in[3];
    const float* gamma = (const float*)d_in[4];
    const float* beta  = (const float*)d_in[5];
    const float* rmean = (const float*)d_in[6];
    const float* rvar  = (const float*)d_in[7];
    const float* W2    = (const float*)d_in[8];
    const float* b2    = (const float*)d_in[9];

    const int N = in_sizes[0] / 128;
    const int E = in_sizes[1] / 2;
    const int* src = eidx;
    const int* dst = eidx + E;

    // ws layout (floats): dinv[N] | scale[128] | shift[128] | g1[N*128] | a1[N*128]
    float* dinv  = (float*)d_ws;
    float* bnscl = dinv + N;
    float* bnsft = bnscl + 128;
    float* g1    = bnsft + 128;
    float* a1    = g1 + (size_t)N * 128;
    float* g2    = g1;                      // g1 dead after edge1; reuse for g2
    float* out   = (float*)d_out;

    const int T = 256;
    const int rowTiles   = (N + 15) / 16;
    const int gemmBlocks = (rowTiles + 7) / 8;   // 8 waves (16-row tiles) / block
    const int edgeBlocks = (E + 7) / 8;          // 8 waves (edges) / block

    // 1) normalization coefficients + folded BN constants
    k_init_deg      <<<(N + T - 1) / T, T, 0, stream>>>(dinv, N);
    k_count_deg     <<<(E + T - 1) / T, T, 0, stream>>>(dst, dinv, E);
    k_rsqrt_inplace <<<(N + T - 1) / T, T, 0, stream>>>(dinv, N);
    k_prep_bn       <<<1, 128, 0, stream>>>(b1, gamma, beta, rmean, rvar, bnscl, bnsft);

    // 2) layer 1 GEMM (WMMA f32, TDM-staged W1) + self-loop seed
    k_gemm1<<<gemmBlocks, T, 0, stream>>>(x, W1, dinv, g1, a1, N);

    // 3) layer 1 edge scatter
    k_edge1<<<edgeBlocks, T, 0, stream>>>(src, dst, dinv, g1, a1, E);

    // 4) layer 2 GEMM (fused bias+BN+ReLU on A-loads) + b2/self-loop seed
    k_gemm2<<<gemmBlocks, T, 0, stream>>>(a1, W2, dinv, bnscl, bnsft, b2, g2, out, N);

    // 5) layer 2 edge scatter into d_out
    k_edge2<<<edgeBlocks, T, 0, stream>>>(src, dst, dinv, g2, out, E);
}